// PointSetAbstractionMsg_31061203485291
// MI455X (gfx1250) — compile-verified
//
#include <hip/hip_runtime.h>
#include <hip/hip_bf16.h>

typedef __attribute__((ext_vector_type(2))) float v2f;
typedef __attribute__((ext_vector_type(8))) float v8f;

#define B_ 8
#define N_ 16384
#define M_ 1024
#define C_ 256

#define KEY_MASK 0xFFFFFC00u   // keep sign+exp+13 mantissa bits; low 10 bits = index

// ---------------------------------------------------------------------------
// Phase 1: transpose centroid_f [B, C, M] -> ftr [B, M, C] (workspace)
// so the per-neighbor feature gather is a coalesced row read.
// ---------------------------------------------------------------------------
__global__ __launch_bounds__(256) void transpose_cf(const float* __restrict__ cf,
                                                    float* __restrict__ ftr) {
  __shared__ float tile[32][33];
  const int b = blockIdx.z;
  const int c0 = blockIdx.x * 32;
  const int m0 = blockIdx.y * 32;
  const int tx = threadIdx.x, ty = threadIdx.y;  // 32 x 8
  const float* src = cf + ((size_t)b * C_ + c0) * M_ + m0;
#pragma unroll
  for (int k = 0; k < 4; ++k)
    tile[ty + 8 * k][tx] = src[(size_t)(ty + 8 * k) * M_ + tx];
  __syncthreads();
  float* dst = ftr + ((size_t)b * M_ + m0) * C_ + c0;
#pragma unroll
  for (int k = 0; k < 4; ++k)
    dst[(size_t)(ty + 8 * k) * C_ + tx] = tile[tx][ty + 8 * k];
}

// Branchless top-3-smallest insert on packed keys:
//   k0' = min(key,k0); k1' = med3(key,k0,k1); k2' = med3(key,k1,k2)
__device__ __forceinline__ void key_insert3(float key, float& k0, float& k1, float& k2) {
  float nk0 = fminf(key, k0);
  float nk1 = __builtin_amdgcn_fmed3f(key, k0, k1);
  float nk2 = __builtin_amdgcn_fmed3f(key, k1, k2);
  k0 = nk0; k1 = nk1; k2 = nk2;
}

// ---------------------------------------------------------------------------
// Phase 2: fused  (WMMA cdist) -> top-3 -> inv-dist weights -> feature interp
// One wave32 owns 16 points; 256-thread block owns 128 points + a 16KB LDS
// copy of this batch's centroids (SoA: cx, cy, cz, |c|^2).
// ---------------------------------------------------------------------------
__global__ __launch_bounds__(256) void psa_interp_kernel(
    const float* __restrict__ xyz, const float* __restrict__ cxyz,
    const float* __restrict__ ftr, float* __restrict__ out) {
  __shared__ float s_cx[M_], s_cy[M_], s_cz[M_], s_c2[M_];
  const int tid = threadIdx.x;
  const int blocksPerBatch = N_ / 128;
  const int b = blockIdx.x / blocksPerBatch;
  const int nblk = blockIdx.x % blocksPerBatch;

  // Stage centroids into LDS (SoA) and precompute |c|^2.
  const float* cb = cxyz + (size_t)b * M_ * 3;
  for (int i = tid; i < M_ * 3; i += 256) {
    float v = cb[i];
    int m = i / 3, comp = i - 3 * m;
    if (comp == 0)      s_cx[m] = v;
    else if (comp == 1) s_cy[m] = v;
    else                s_cz[m] = v;
  }
  __syncthreads();
  for (int m = tid; m < M_; m += 256) {
    float x = s_cx[m], y = s_cy[m], z = s_cz[m];
    s_c2[m] = x * x + y * y + z * z;
  }
  __syncthreads();

  const int lane = tid & 31;
  const int wave = tid >> 5;
  const int col  = lane & 15;          // point column within the tile
  const bool hi  = lane >= 16;
  const int n0   = nblk * 128 + wave * 16;  // first of this wave's 16 points

  // This lane's point (lanes p and p+16 share point p).
  const float* xp = xyz + ((size_t)b * N_ + (n0 + col)) * 3;
  const float px = xp[0], py = xp[1], pz = xp[2];
  const float x2 = px * px + py * py + pz * pz;

  // B operand (4x16): rows are (-2x, -2y, -2z, 1); VGPR0 = K0|K2, VGPR1 = K1|K3.
  v2f bv;
  bv[0] = hi ? -2.0f * pz : -2.0f * px;
  bv[1] = hi ?  1.0f      : -2.0f * py;

  // A operand (16x4): rows are centroids, cols (cx, cy, cz, |c|^2).
  const float* pa0 = (hi ? s_cz : s_cx) + col;
  const float* pa1 = (hi ? s_c2 : s_cy) + col;
  const int rowoff = hi ? 8 : 0;       // D rows v+8 land in hi lanes

  // Packed (d^2 | index) keys, initialized to +huge.
  float k0 = __uint_as_float(__float_as_uint(1e30f) & KEY_MASK);
  float k1 = k0, k2 = k0;

#pragma unroll 4
  for (int cbase = 0; cbase < M_; cbase += 16) {
    v2f av;
    av[0] = pa0[cbase];
    av[1] = pa1[cbase];
    v8f acc;
#pragma unroll
    for (int t = 0; t < 8; ++t) acc[t] = x2;   // C = |x|^2 (per column)
    // D = A*B + C = |c|^2 - 2 c.x + |x|^2  (squared distance tile)
    acc = __builtin_amdgcn_wmma_f32_16x16x4_f32(
        false, av, false, bv, (short)0, acc, false, false);
    const unsigned base = (unsigned)(cbase + rowoff);
#pragma unroll
    for (int t = 0; t < 8; ++t) {
      float key = __uint_as_float((__float_as_uint(acc[t]) & KEY_MASK) | (base + t));
      key_insert3(key, k0, k1, k2);
    }
  }

  // Merge the two half-wave top-3 lists (partner lane ^ 16).
  {
    float e0 = __shfl_xor(k0, 16, 32);
    float e1 = __shfl_xor(k1, 16, 32);
    float e2 = __shfl_xor(k2, 16, 32);
    key_insert3(e0, k0, k1, k2);
    key_insert3(e1, k0, k1, k2);
    key_insert3(e2, k0, k1, k2);
  }

  // Unpack keys -> (index, squared distance).
  const unsigned u0 = __float_as_uint(k0);
  const unsigned u1 = __float_as_uint(k1);
  const unsigned u2 = __float_as_uint(k2);
  const int i0 = (int)(u0 & 1023u), i1 = (int)(u1 & 1023u), i2 = (int)(u2 & 1023u);
  const float d0 = __uint_as_float(u0 & KEY_MASK);
  const float d1 = __uint_as_float(u1 & KEY_MASK);
  const float d2 = __uint_as_float(u2 & KEY_MASK);

  // Inverse-distance weights (on sqrt distances, like the reference).
  float t0 = sqrtf(fmaxf(d0, 0.0f));
  float t1 = sqrtf(fmaxf(d1, 0.0f));
  float t2 = sqrtf(fmaxf(d2, 0.0f));
  float w0 = 1.0f / fmaxf(t0, 1e-8f);
  float w1 = 1.0f / fmaxf(t1, 1e-8f);
  float w2 = 1.0f / fmaxf(t2, 1e-8f);
  const float wsum = w0 + w1 + w2;
  w0 /= wsum; w1 /= wsum; w2 /= wsum;

  // Weighted feature gather: 32 lanes x 8 channels cover C=256 per point.
  const int ch = lane * 8;
  const float* fb = ftr + (size_t)b * M_ * C_;
  float* ob = out + ((size_t)b * N_ + n0) * C_;
#pragma unroll 1
  for (int q = 0; q < 16; ++q) {
    const int   j0 = __shfl(i0, q, 32), j1 = __shfl(i1, q, 32), j2 = __shfl(i2, q, 32);
    const float u0w = __shfl(w0, q, 32), u1w = __shfl(w1, q, 32), u2w = __shfl(w2, q, 32);
    const float4* r0 = (const float4*)(fb + (size_t)j0 * C_ + ch);
    const float4* r1 = (const float4*)(fb + (size_t)j1 * C_ + ch);
    const float4* r2 = (const float4*)(fb + (size_t)j2 * C_ + ch);
    float4 a0 = r0[0], a1 = r0[1];
    float4 b0 = r1[0], b1 = r1[1];
    float4 c0 = r2[0], c1 = r2[1];
    float4 o0, o1;
    o0.x = u0w * a0.x + u1w * b0.x + u2w * c0.x;
    o0.y = u0w * a0.y + u1w * b0.y + u2w * c0.y;
    o0.z = u0w * a0.z + u1w * b0.z + u2w * c0.z;
    o0.w = u0w * a0.w + u1w * b0.w + u2w * c0.w;
    o1.x = u0w * a1.x + u1w * b1.x + u2w * c1.x;
    o1.y = u0w * a1.y + u1w * b1.y + u2w * c1.y;
    o1.z = u0w * a1.z + u1w * b1.z + u2w * c1.z;
    o1.w = u0w * a1.w + u1w * b1.w + u2w * c1.w;
    float4* op = (float4*)(ob + (size_t)q * C_ + ch);
    op[0] = o0;
    op[1] = o1;
  }
}

extern "C" void kernel_launch(void* const* d_in, const int* in_sizes, int n_in,
                              void* d_out, int out_size, void* d_ws, size_t ws_size,
                              hipStream_t stream) {
  (void)in_sizes; (void)n_in; (void)out_size; (void)ws_size;
  const float* xyz  = (const float*)d_in[0];  // [B, N, 3]
  const float* cxyz = (const float*)d_in[1];  // [B, M, 3]
  const float* cf   = (const float*)d_in[2];  // [B, C, M]
  float* out = (float*)d_out;                 // [B, N, C]
  float* ftr = (float*)d_ws;                  // [B, M, C] = 8 MiB scratch

  dim3 tb(32, 8, 1);
  dim3 tg(C_ / 32, M_ / 32, B_);
  transpose_cf<<<tg, tb, 0, stream>>>(cf, ftr);

  const int grid = B_ * (N_ / 128);  // 8 waves/block, 16 points/wave
  psa_interp_kernel<<<grid, 256, 0, stream>>>(xyz, cxyz, ftr, out);
}